// OpeningLoss2D_47107201302668
// MI455X (gfx1250) — compile-verified
//
#include <hip/hip_runtime.h>
#include <hip/hip_bf16.h>

// Problem constants (reference: labels [16,8,512,512] f32, scalar f32 out)
#define NB      16
#define NC      8
#define H       512
#define W       512
#define PLANES  (NB * NC)          // 128
#define ROWS    32                 // output rows per block
#define STRIPS  (H / ROWS)         // 16
#define NBLOCKS (PLANES * STRIPS)  // 2048
#define PITCH   516                // LDS row pitch in floats (16B-aligned, bank-skewed)
#define TROWS   (ROWS + 2)         // staged rows: y0-1 .. y0+ROWS
#define NTHREADS 256

typedef float v2f __attribute__((ext_vector_type(2)));
typedef float v8f __attribute__((ext_vector_type(8)));
typedef int   v4i __attribute__((ext_vector_type(4)));

#if defined(__gfx1250__) && __has_builtin(__builtin_amdgcn_global_load_async_to_lds_b128)
#define USE_ASYNC 1
#else
#define USE_ASYNC 0
#endif

__device__ __forceinline__ float wave_reduce_wmma(float acc) {
    // D = A x B + 0 with B = ones: D[m,n] = acc[m] + acc[m+16] (A 16x4 layout:
    // lanes 0-15 carry K=0/1, lanes 16-31 carry K=2/3; we put acc in K=0/2, 0 in K=1/3).
    v2f a; a[0] = acc; a[1] = 0.0f;
    v2f b; b[0] = 1.0f; b[1] = 1.0f;
    v8f c = {};
    v8f d = __builtin_amdgcn_wmma_f32_16x16x4_f32(false, a, false, b, (short)0, c, false, false);
    float s = 0.0f;
#pragma unroll
    for (int i = 0; i < 8; ++i) s += d[i];
    // lane L (0..15): sum over M=0..7 of column L ; lane L+16: M=8..15.
    // s(L) + s(L+16) == total over all 32 lanes.
    return s;
}

__device__ __forceinline__ void block_reduce_store(float acc, float* out, float scale,
                                                   float* red /* >=16 floats LDS */) {
    float s = wave_reduce_wmma(acc);
    const int tid  = threadIdx.x;
    const int lane = tid & 31;
    const int wid  = tid >> 5;
    if ((lane & 15) == 0) red[wid * 2 + (lane >> 4)] = s;
    __syncthreads();
    if (tid == 0) {
        float t = 0.0f;
#pragma unroll
        for (int i = 0; i < 16; ++i) t += red[i];  // 8 waves x 2 entries
        *out = t * scale;
    }
}

__global__ __launch_bounds__(NTHREADS)
void opening_mse_partial(const float* __restrict__ labels, float* __restrict__ partial) {
    __shared__ float tile[TROWS * PITCH];
    __shared__ float red[16];

    const int bid   = blockIdx.x;
    const int plane = bid / STRIPS;
    const int strip = bid % STRIPS;
    const int y0    = strip * ROWS;
    const int tid   = threadIdx.x;
    const float* pbase = labels + (size_t)plane * H * W;

    // ---- Stage rows [y0-1 .. y0+ROWS] (clamped = symmetric pad for size-2) into LDS ----
#pragma unroll
    for (int t = tid; t < TROWS * (W / 4); t += NTHREADS) {
        const int row = t >> 7;        // 0 .. TROWS-1
        const int c4  = t & 127;       // float4 index within row
        int gr = y0 - 1 + row;
        gr = gr < 0 ? 0 : (gr > H - 1 ? H - 1 : gr);
        const float* src = pbase + gr * W + c4 * 4;
        float*       dst = &tile[row * PITCH + c4 * 4];
#if USE_ASYNC
        __builtin_amdgcn_global_load_async_to_lds_b128(
            (__attribute__((address_space(1))) v4i*)src,
            (__attribute__((address_space(3))) v4i*)dst, 0, 0);
#else
        *(float4*)dst = *(const float4*)src;
#endif
    }
#if USE_ASYNC
#if __has_builtin(__builtin_amdgcn_s_wait_asynccnt)
    __builtin_amdgcn_s_wait_asynccnt(0);
#else
    asm volatile("s_wait_asynccnt 0" ::: "memory");
#endif
#endif
    __syncthreads();

    // ---- Compute: erosion(2x2, pad low) then dilation(2x2, pad high), accumulate MSE ----
    float acc = 0.0f;
#pragma unroll
    for (int it = 0; it < (ROWS * (W / 4)) / NTHREADS; ++it) {
        const int slot = tid + it * NTHREADS;
        const int yrow = slot >> 7;      // 0..ROWS-1
        const int c4   = slot & 127;
        const int y    = y0 + yrow;
        const int x    = c4 * 4;

        // LDS row indices (tile row 0 == global row y0-1, clamped at stage time)
        const int rt = yrow;            // global y-1 (low clamp handled by staging)
        const int rm = yrow + 1;        // global y
        const int r1 = (y + 1 < H) ? (y + 1) : (H - 1);  // eroded-row clamp (dilation)
        const int rq = r1 - y0 + 1;     // LDS row of r1
        const int rp = rq - 1;          // LDS row of r1-1

        // 6 columns: x-1(clamped), x..x+3, x+4(clamped)
        int col[6];
        col[0] = (x > 0) ? x - 1 : 0;
        col[1] = x; col[2] = x + 1; col[3] = x + 2; col[4] = x + 3;
        col[5] = (x + 4 < W) ? x + 4 : (W - 1);

        float A[6], Bv[6], P[6], Q[6];
#pragma unroll
        for (int k = 0; k < 6; ++k) {
            A[k]  = tile[rt * PITCH + col[k]];
            Bv[k] = tile[rm * PITCH + col[k]];
            P[k]  = tile[rp * PITCH + col[k]];
            Q[k]  = tile[rq * PITCH + col[k]];
        }

        float m1[6], m2[6];
#pragma unroll
        for (int k = 0; k < 6; ++k) {
            m1[k] = fminf(A[k], Bv[k]);   // column mins for eroded row y
            m2[k] = fminf(P[k], Q[k]);    // column mins for eroded row min(y+1,H-1)
        }

        float em[5];
#pragma unroll
        for (int t2 = 0; t2 < 5; ++t2) {
            const float eY = fminf(m1[t2], m1[t2 + 1]);
            const float eR = fminf(m2[t2], m2[t2 + 1]);
            em[t2] = fmaxf(eY, eR);       // max over the two eroded rows
        }
        if (x + 4 > W - 1) em[4] = em[3]; // high-side column clamp of dilation

#pragma unroll
        for (int k = 0; k < 4; ++k) {
            const float opened = fmaxf(em[k], em[k + 1]);
            const float lab    = Bv[k + 1];
            const float df     = lab - opened;
            acc = fmaf(df, df, acc);
        }
    }
    __syncthreads();

    block_reduce_store(acc, &partial[bid], 1.0f, red);
}

__global__ __launch_bounds__(NTHREADS)
void opening_mse_final(const float* __restrict__ partial, float* __restrict__ out) {
    __shared__ float red[16];
    float acc = 0.0f;
    for (int i = threadIdx.x; i < NBLOCKS; i += NTHREADS) acc += partial[i];
    const float invN = 0x1p-25f;  // 1 / (16*8*512*512)
    block_reduce_store(acc, out, invN, red);
}

extern "C" void kernel_launch(void* const* d_in, const int* in_sizes, int n_in,
                              void* d_out, int out_size, void* d_ws, size_t ws_size,
                              hipStream_t stream) {
    const float* labels = (const float*)d_in[0];
    float* partial = (float*)d_ws;       // NBLOCKS floats of scratch
    float* out     = (float*)d_out;      // 1 float
    (void)in_sizes; (void)n_in; (void)out_size; (void)ws_size;

    opening_mse_partial<<<NBLOCKS, NTHREADS, 0, stream>>>(labels, partial);
    opening_mse_final<<<1, NTHREADS, 0, stream>>>(partial, out);
}